// SlotAttention_42829413875859
// MI455X (gfx1250) — compile-verified
//
#include <hip/hip_runtime.h>
#include <hip/hip_bf16.h>

#define B_ 32
#define S_ 11
#define R_ 16
#define D_ 64
#define C_ 512
#define N_ 256
#define H_ 128
#define NUM_ITER_ 3

typedef float v2f __attribute__((ext_vector_type(2)));
typedef float v8f __attribute__((ext_vector_type(8)));

__device__ __forceinline__ float grid_coord(int i) {
  return -1.0f + (2.0f / 15.0f) * (float)i;
}

// C += A(16x16 of As, row-major, K contiguous) * B(16x16 of Bp), fp32 WMMA.
// A fragment (ISA 7.12.2): lane%16 = row M, lane/16 selects K pair {0,1}/{2,3}.
// B is stored K-PAIR-SWIZZLED in LDS: Bp[((k>>1)*ldn + n)*2 + (k&1)] = W[k][n],
// so each B fragment is one aligned 8-byte ds_load_b64 (matches the A side).
__device__ __forceinline__ v8f wmma_tile_f32(const float* As, int lda, int m0,
                                             const float* Bp, int ldn, int n0,
                                             int K, v8f acc, int lane) {
  const int mn = lane & 15;
  const int kk = (lane >> 4) << 1;
  for (int k0 = 0; k0 < K; k0 += 4) {
    v2f a = *(const v2f*)&As[(m0 + mn) * lda + k0 + kk];
    v2f b = *(const v2f*)&Bp[(((k0 + kk) >> 1) * ldn + n0 + mn) * 2];
    acc = __builtin_amdgcn_wmma_f32_16x16x4_f32(false, a, false, b, (short)0,
                                                acc, false, false);
  }
  return acc;
}

// Swizzled LDS store helper for a K x N row-major weight block.
__device__ __forceinline__ void store_w_swz(float* Wp, int ldn, int k, int n,
                                            float w) {
  Wp[(((k >> 1) * ldn + n) << 1) + (k & 1)] = w;
}

// ---------------------------------------------------------------- pos embed
__global__ __launch_bounds__(128) void pos_embed_kernel(
    const float* __restrict__ es_w1, const float* __restrict__ es_b1,
    const float* __restrict__ es_w2, const float* __restrict__ es_b2,
    float* __restrict__ pos) {
  __shared__ float h[128];
  const int n = blockIdx.x;
  const int t = threadIdx.x;
  const float gx = grid_coord(n >> 4), gy = grid_coord(n & 15);
  h[t] = fmaxf(gx * es_w1[t] + gy * es_w1[128 + t] + es_b1[t], 0.0f);
  __syncthreads();
  for (int c = t; c < C_; c += 128) {
    float acc = es_b2[c];
    for (int j = 0; j < 128; ++j) acc += h[j] * es_w2[j * C_ + c];
    pos[n * C_ + c] = acc;
  }
}

// ------------------------------------------- inputs_k / inputs_v projection
// 64 output rows per block; K tiled by 64; 8 waves own 4 (16x16) tiles each
// (16 tiles for the k projection, 16 for v).
__global__ __launch_bounds__(256) void proj_kv_kernel(
    const float* __restrict__ inputs, const float* __restrict__ pos,
    const float* __restrict__ pk_w, const float* __restrict__ pk_b,
    const float* __restrict__ pv_w, const float* __restrict__ pv_b,
    float* __restrict__ ink, float* __restrict__ inv) {
  __shared__ float Xs[64 * 64];
  __shared__ float Wk[64 * 64];   // K-pair swizzled
  __shared__ float Wv[64 * 64];   // K-pair swizzled
  const int t = threadIdx.x;
  const int wave = t >> 5, lane = t & 31;
  const int row0 = blockIdx.x * 64;
  v8f acc[4];
  const v8f vzero = {0.f, 0.f, 0.f, 0.f, 0.f, 0.f, 0.f, 0.f};
  for (int i = 0; i < 4; ++i) acc[i] = vzero;

  for (int kc = 0; kc < C_; kc += 64) {
    for (int idx = t; idx < 64 * 64; idx += 256) {
      int r = idx >> 6, c = idx & 63;
      int grow = row0 + r;
      Xs[idx] = inputs[(size_t)grow * C_ + kc + c] +
                pos[(grow & (N_ - 1)) * C_ + kc + c];
      store_w_swz(Wk, 64, r, c, pk_w[(kc + r) * D_ + c]);
      store_w_swz(Wv, 64, r, c, pv_w[(kc + r) * D_ + c]);
    }
    __syncthreads();
    for (int tt = 0; tt < 4; ++tt) {
      int tile = wave * 4 + tt;     // 0..31 ; <16 -> k, >=16 -> v
      int isv = tile >> 4;
      int lt = tile & 15;
      int m0 = (lt >> 2) * 16, n0 = (lt & 3) * 16;
      const float* Bs = isv ? Wv : Wk;
      acc[tt] = wmma_tile_f32(Xs, 64, m0, Bs, 64, n0, 64, acc[tt], lane);
    }
    __syncthreads();
  }
  const int col = lane & 15, rb = (lane >> 4) << 3;
  for (int tt = 0; tt < 4; ++tt) {
    int tile = wave * 4 + tt;
    int isv = tile >> 4;
    int lt = tile & 15;
    int m0 = (lt >> 2) * 16, n0 = (lt & 3) * 16;
    float bias = isv ? pv_b[n0 + col] : pk_b[n0 + col];
    float* out = isv ? inv : ink;
    for (int i = 0; i < 8; ++i)
      out[(size_t)(row0 + m0 + rb + i) * D_ + n0 + col] = acc[tt][i] + bias;
  }
}

// --------------------------------------------------------------- state init
__global__ void init_state_kernel(const float* __restrict__ slots_mu,
                                  const float* __restrict__ s_p0,
                                  const float* __restrict__ s_s0,
                                  float* __restrict__ slots,
                                  float* __restrict__ s_p,
                                  float* __restrict__ s_s) {
  int idx = blockIdx.x * blockDim.x + threadIdx.x;
  if (idx < B_ * S_ * D_) slots[idx] = slots_mu[idx % (S_ * D_)];
  if (idx < B_ * S_ * 2) { s_p[idx] = s_p0[idx]; s_s[idx] = s_s0[idx]; }
}

// --------------------------------------------------- q = LN(slots)@pq / 8
__global__ __launch_bounds__(64) void q_proj_kernel(
    const float* __restrict__ slots, const float* __restrict__ ns_g,
    const float* __restrict__ ns_b, const float* __restrict__ pq_w,
    const float* __restrict__ pq_b, float* __restrict__ q) {
  __shared__ float red[64];
  __shared__ float xn[64];
  const int bs = blockIdx.x, d = threadIdx.x;
  float x = slots[bs * D_ + d];
  red[d] = x; __syncthreads();
  for (int off = 32; off > 0; off >>= 1) {
    if (d < off) red[d] += red[d + off];
    __syncthreads();
  }
  float m = red[0] * (1.0f / D_);
  __syncthreads();
  red[d] = (x - m) * (x - m); __syncthreads();
  for (int off = 32; off > 0; off >>= 1) {
    if (d < off) red[d] += red[d + off];
    __syncthreads();
  }
  float var = red[0] * (1.0f / D_);
  xn[d] = (x - m) * rsqrtf(var + 1e-3f) * ns_g[d] + ns_b[d];
  __syncthreads();
  float acc = pq_b[d];
  for (int e = 0; e < D_; ++e) acc += xn[e] * pq_w[e * D_ + d];
  q[bs * D_ + d] = acc * 0.125f;   // D^-0.5
}

// ------------------------------------------- k/v = ff(in + pe), fused logits
// One block per (b, s, 32-row n-tile). Phase 0: k stream -> logits only.
// Phase 1: v stream -> global v. LDS: W(32KB) + T(8KB) + Hd(16KB) = 56KB.
__global__ __launch_bounds__(256) void kv_attn_kernel(
    const float* __restrict__ ink, const float* __restrict__ inv,
    const float* __restrict__ s_p, const float* __restrict__ s_s,
    const float* __restrict__ er_w, const float* __restrict__ er_b,
    const float* __restrict__ mi_w1, const float* __restrict__ mi_b1,
    const float* __restrict__ mi_w2, const float* __restrict__ mi_b2,
    const float* __restrict__ q, float* __restrict__ v_out,
    float* __restrict__ logits) {
  __shared__ float Wsh[D_ * H_];   // 8192 f : mi_w1, then mi_w2 (swizzled)
  __shared__ float Tt[32 * D_];    // 2048 f : input tile, reused as k tile
  __shared__ float Hd[32 * H_];    // 4096 f : hidden
  __shared__ float b1s[H_];
  __shared__ float b2s[D_];
  __shared__ float qs[D_];
  const int t = threadIdx.x;
  const int wave = t >> 5, lane = t & 31;
  const int ntile = blockIdx.x & 7;
  const int bs = blockIdx.x >> 3;
  const int b = bs / S_, s = bs - b * S_;
  const int n0 = ntile * 32;
  const v8f vzero = {0.f, 0.f, 0.f, 0.f, 0.f, 0.f, 0.f, 0.f};

  if (t < H_) b1s[t] = mi_b1[t];
  if (t < D_) { b2s[t] = mi_b2[t]; qs[t] = q[bs * D_ + t]; }
  const float px = s_p[bs * 2 + 0], py = s_p[bs * 2 + 1];
  const float sx = s_s[bs * 2 + 0], sy = s_s[bs * 2 + 1];
  const int col = lane & 15, rb = (lane >> 4) << 3;

  for (int phase = 0; phase < 2; ++phase) {
    const float* inb = phase ? inv : ink;
    // build T = in + relative positional encoding
    for (int idx = t; idx < 32 * D_; idx += 256) {
      int r = idx >> 6, dd = idx & 63;
      int n = n0 + r;
      float rx = (grid_coord(n >> 4) - px) / sx;
      float ry = (grid_coord(n & 15) - py) / sy;
      float pe = rx * er_w[dd] + ry * er_w[D_ + dd] + er_b[dd];
      Tt[idx] = inb[((size_t)b * N_ + n) * D_ + dd] + pe;
    }
    for (int idx = t; idx < D_ * H_; idx += 256)
      store_w_swz(Wsh, H_, idx >> 7, idx & 127, mi_w1[idx]);
    __syncthreads();

    // GEMM1: [32x64]@[64x128] -> relu -> Hd.  16 tiles, 2 per wave.
    for (int tt = 0; tt < 2; ++tt) {
      int tile = wave * 2 + tt;
      int m0 = (tile >> 3) * 16, nn0 = (tile & 7) * 16;
      v8f acc = vzero;
      acc = wmma_tile_f32(Tt, D_, m0, Wsh, H_, nn0, D_, acc, lane);
      float bias = b1s[nn0 + col];
      for (int i = 0; i < 8; ++i)
        Hd[(m0 + rb + i) * H_ + nn0 + col] = fmaxf(acc[i] + bias, 0.0f);
    }
    __syncthreads();
    for (int idx = t; idx < H_ * D_; idx += 256)
      store_w_swz(Wsh, D_, idx >> 6, idx & 63, mi_w2[idx]);
    __syncthreads();

    // GEMM2: [32x128]@[128x64].  8 tiles, 1 per wave.
    {
      int m0 = (wave >> 2) * 16, nn0 = (wave & 3) * 16;
      v8f acc = vzero;
      acc = wmma_tile_f32(Hd, H_, m0, Wsh, D_, nn0, H_, acc, lane);
      float bias = b2s[nn0 + col];
      if (phase == 0) {
        for (int i = 0; i < 8; ++i)
          Tt[(m0 + rb + i) * D_ + nn0 + col] = acc[i] + bias;   // k tile
      } else {
        for (int i = 0; i < 8; ++i)
          v_out[((size_t)bs * N_ + n0 + m0 + rb + i) * D_ + nn0 + col] =
              acc[i] + bias;
      }
    }
    __syncthreads();
    if (phase == 0 && t < 32) {     // logits[b, n, s] = k . q
      float accl = 0.0f;
      for (int dd = 0; dd < D_; ++dd) accl += Tt[t * D_ + dd] * qs[dd];
      logits[((size_t)b * N_ + n0 + t) * S_ + s] = accl;
    }
    __syncthreads();
  }
}

// ------------------------------------------------------ softmax over slots
__global__ void softmax_kernel(const float* __restrict__ logits,
                               float* __restrict__ attn) {
  int idx = blockIdx.x * blockDim.x + threadIdx.x;
  if (idx >= B_ * N_) return;
  float l[S_];
  float mx = -3.402823466e38f;
  for (int s = 0; s < S_; ++s) {
    l[s] = logits[(size_t)idx * S_ + s];
    mx = fmaxf(mx, l[s]);
  }
  float sum = 0.0f;
  for (int s = 0; s < S_; ++s) { l[s] = expf(l[s] - mx); sum += l[s]; }
  float inv = 1.0f / sum;
  for (int s = 0; s < S_; ++s)
    attn[(size_t)idx * S_ + s] = l[s] * inv + 1e-8f;
}

// ------------------------------------------------- column sums over inputs
__global__ __launch_bounds__(256) void colsum_kernel(
    const float* __restrict__ attn, float* __restrict__ colsum) {
  __shared__ float red[256];
  const int bs = blockIdx.x, b = bs / S_, s = bs - (bs / S_) * S_;
  const int n = threadIdx.x;
  red[n] = attn[((size_t)b * N_ + n) * S_ + s];
  __syncthreads();
  for (int off = 128; off > 0; off >>= 1) {
    if (n < off) red[n] += red[n + off];
    __syncthreads();
  }
  if (n == 0) colsum[bs] = red[0];
}

// ------------------------------------------------- updates = attn_norm . v
__global__ __launch_bounds__(64) void updates_kernel(
    const float* __restrict__ attn, const float* __restrict__ colsum,
    const float* __restrict__ v, float* __restrict__ updates) {
  __shared__ float a_sh[N_];
  const int bs = blockIdx.x, b = bs / S_, s = bs - (bs / S_) * S_;
  const int d = threadIdx.x;
  for (int n = d; n < N_; n += 64)
    a_sh[n] = attn[((size_t)b * N_ + n) * S_ + s];
  __syncthreads();
  float inv = 1.0f / colsum[bs];
  float acc = 0.0f;
  for (int n = 0; n < N_; ++n)
    acc += a_sh[n] * v[((size_t)bs * N_ + n) * D_ + d];
  updates[bs * D_ + d] = acc * inv;
}

// ------------------------------------------------------------ calcSP stage
__global__ __launch_bounds__(256) void calc_sp_kernel(
    const float* __restrict__ attn, const float* __restrict__ colsum,
    float* __restrict__ s_p, float* __restrict__ s_s) {
  __shared__ float r1[256], r2[256];
  __shared__ float wp[2];
  const int bs = blockIdx.x, b = bs / S_, s = bs - (bs / S_) * S_;
  const int n = threadIdx.x;
  float a = attn[((size_t)b * N_ + n) * S_ + s] / colsum[bs];
  float gx = grid_coord(n >> 4), gy = grid_coord(n & 15);
  r1[n] = a * gx; r2[n] = a * gy;
  __syncthreads();
  for (int off = 128; off > 0; off >>= 1) {
    if (n < off) { r1[n] += r1[n + off]; r2[n] += r2[n + off]; }
    __syncthreads();
  }
  if (n == 0) { wp[0] = r1[0]; wp[1] = r2[0]; }
  __syncthreads();
  float wpx = wp[0], wpy = wp[1];
  float aw = a + 1e-11f;
  r1[n] = (gx - wpx) * (gx - wpx) * aw;
  r2[n] = (gy - wpy) * (gy - wpy) * aw;
  __syncthreads();
  for (int off = 128; off > 0; off >>= 1) {
    if (n < off) { r1[n] += r1[n + off]; r2[n] += r2[n + off]; }
    __syncthreads();
  }
  if (n == 0) {
    s_p[bs * 2 + 0] = wpx;
    s_p[bs * 2 + 1] = wpy;
    s_s[bs * 2 + 0] = fminf(fmaxf(sqrtf(r1[0]), 0.001f), 5.0f);
    s_s[bs * 2 + 1] = fminf(fmaxf(sqrtf(r2[0]), 0.001f), 5.0f);
  }
}

// --------------------------------------------------- GRU + residual MLP
__global__ __launch_bounds__(64) void gru_mlp_kernel(
    const float* __restrict__ updates, const float* __restrict__ gru_k,
    const float* __restrict__ gru_rk, const float* __restrict__ gru_b,
    const float* __restrict__ nm_g, const float* __restrict__ nm_b,
    const float* __restrict__ mlp_w1, const float* __restrict__ mlp_b1,
    const float* __restrict__ mlp_w2, const float* __restrict__ mlp_b2,
    float* __restrict__ slots) {
  __shared__ float u[64], h[64], xn[64], hid[128], red[64];
  const int bs = blockIdx.x, d = threadIdx.x;
  u[d] = updates[bs * D_ + d];
  h[d] = slots[bs * D_ + d];
  __syncthreads();
  float xz = gru_b[d], xr = gru_b[64 + d], xh = gru_b[128 + d];
  float hz = gru_b[192 + d], hr = gru_b[256 + d], hh = gru_b[320 + d];
  for (int e = 0; e < D_; ++e) {
    float ue = u[e], he = h[e];
    const float* kr = gru_k + e * 192;
    const float* rr = gru_rk + e * 192;
    xz += ue * kr[d];      xr += ue * kr[64 + d];  xh += ue * kr[128 + d];
    hz += he * rr[d];      hr += he * rr[64 + d];  hh += he * rr[128 + d];
  }
  float z = 1.0f / (1.0f + expf(-(xz + hz)));
  float r = 1.0f / (1.0f + expf(-(xr + hr)));
  float hc = tanhf(xh + r * hh);
  float sn = z * h[d] + (1.0f - z) * hc;
  // LayerNorm (nm)
  red[d] = sn; __syncthreads();
  for (int off = 32; off > 0; off >>= 1) {
    if (d < off) red[d] += red[d + off];
    __syncthreads();
  }
  float m = red[0] * (1.0f / D_);
  __syncthreads();
  red[d] = (sn - m) * (sn - m); __syncthreads();
  for (int off = 32; off > 0; off >>= 1) {
    if (d < off) red[d] += red[d + off];
    __syncthreads();
  }
  float var = red[0] * (1.0f / D_);
  xn[d] = (sn - m) * rsqrtf(var + 1e-3f) * nm_g[d] + nm_b[d];
  __syncthreads();
  for (int j = d; j < H_; j += 64) {
    float a = mlp_b1[j];
    for (int e = 0; e < D_; ++e) a += xn[e] * mlp_w1[e * H_ + j];
    hid[j] = fmaxf(a, 0.0f);
  }
  __syncthreads();
  float o = mlp_b2[d];
  for (int j = 0; j < H_; ++j) o += hid[j] * mlp_w2[j * D_ + d];
  slots[bs * D_ + d] = sn + o;
}

// -------------------------------------------------------------- finalize
__global__ void finalize_kernel(const float* __restrict__ slots,
                                const float* __restrict__ s_p,
                                const float* __restrict__ s_s,
                                const float* __restrict__ attn,
                                const float* __restrict__ colsum,
                                float* __restrict__ out) {
  int idx = blockIdx.x * blockDim.x + threadIdx.x;
  const int n_slots = B_ * S_ * D_;   // 22528
  const int n_ps = B_ * S_ * 2;       // 704
  const int n_attn = B_ * N_ * S_;    // 90112
  if (idx < n_slots) { out[idx] = slots[idx]; return; }
  idx -= n_slots;
  if (idx < n_ps) { out[n_slots + idx] = s_p[idx]; return; }
  idx -= n_ps;
  if (idx < n_ps) { out[n_slots + n_ps + idx] = s_s[idx]; return; }
  idx -= n_ps;
  if (idx < n_attn) {
    int bn = idx / S_, s = idx - bn * S_;
    int b = bn >> 8;
    out[n_slots + 2 * n_ps + idx] = attn[idx] / colsum[b * S_ + s];
  }
}

extern "C" void kernel_launch(void* const* d_in, const int* in_sizes, int n_in,
                              void* d_out, int out_size, void* d_ws,
                              size_t ws_size, hipStream_t stream) {
  (void)in_sizes; (void)n_in; (void)out_size; (void)ws_size;
  const float* inputs   = (const float*)d_in[0];
  const float* s_p0     = (const float*)d_in[1];
  const float* s_s0     = (const float*)d_in[2];
  const float* slots_mu = (const float*)d_in[3];
  const float* es_w1    = (const float*)d_in[4];
  const float* es_b1    = (const float*)d_in[5];
  const float* es_w2    = (const float*)d_in[6];
  const float* es_b2    = (const float*)d_in[7];
  const float* er_w     = (const float*)d_in[8];
  const float* er_b     = (const float*)d_in[9];
  const float* pk_w     = (const float*)d_in[10];
  const float* pk_b     = (const float*)d_in[11];
  const float* pv_w     = (const float*)d_in[12];
  const float* pv_b     = (const float*)d_in[13];
  const float* pq_w     = (const float*)d_in[14];
  const float* pq_b     = (const float*)d_in[15];
  const float* mi_w1    = (const float*)d_in[16];
  const float* mi_b1    = (const float*)d_in[17];
  const float* mi_w2    = (const float*)d_in[18];
  const float* mi_b2    = (const float*)d_in[19];
  const float* ns_g     = (const float*)d_in[20];
  const float* ns_b     = (const float*)d_in[21];
  const float* nm_g     = (const float*)d_in[22];
  const float* nm_b     = (const float*)d_in[23];
  const float* gru_k    = (const float*)d_in[24];
  const float* gru_rk   = (const float*)d_in[25];
  const float* gru_b    = (const float*)d_in[26];
  const float* mlp_w1   = (const float*)d_in[27];
  const float* mlp_b1   = (const float*)d_in[28];
  const float* mlp_w2   = (const float*)d_in[29];
  const float* mlp_b2   = (const float*)d_in[30];

  float* ws = (float*)d_ws;
  float* pos     = ws;  ws += (size_t)N_ * C_;            // 131072
  float* ink     = ws;  ws += (size_t)B_ * N_ * D_;       // 524288
  float* inv     = ws;  ws += (size_t)B_ * N_ * D_;       // 524288
  float* v       = ws;  ws += (size_t)B_ * S_ * N_ * D_;  // 5767168
  float* q       = ws;  ws += (size_t)B_ * S_ * D_;
  float* logits  = ws;  ws += (size_t)B_ * N_ * S_;
  float* attn    = ws;  ws += (size_t)B_ * N_ * S_;
  float* colsum  = ws;  ws += (size_t)B_ * S_;
  float* updates = ws;  ws += (size_t)B_ * S_ * D_;
  float* slots   = ws;  ws += (size_t)B_ * S_ * D_;
  float* s_p     = ws;  ws += (size_t)B_ * S_ * 2;
  float* s_s     = ws;  ws += (size_t)B_ * S_ * 2;

  pos_embed_kernel<<<N_, 128, 0, stream>>>(es_w1, es_b1, es_w2, es_b2, pos);
  proj_kv_kernel<<<(B_ * N_) / 64, 256, 0, stream>>>(inputs, pos, pk_w, pk_b,
                                                     pv_w, pv_b, ink, inv);
  init_state_kernel<<<(B_ * S_ * D_ + 255) / 256, 256, 0, stream>>>(
      slots_mu, s_p0, s_s0, slots, s_p, s_s);

  for (int ind = 0; ind <= NUM_ITER_; ++ind) {
    q_proj_kernel<<<B_ * S_, 64, 0, stream>>>(slots, ns_g, ns_b, pq_w, pq_b, q);
    kv_attn_kernel<<<B_ * S_ * 8, 256, 0, stream>>>(
        ink, inv, s_p, s_s, er_w, er_b, mi_w1, mi_b1, mi_w2, mi_b2, q, v,
        logits);
    softmax_kernel<<<(B_ * N_) / 256, 256, 0, stream>>>(logits, attn);
    colsum_kernel<<<B_ * S_, 256, 0, stream>>>(attn, colsum);
    calc_sp_kernel<<<B_ * S_, 256, 0, stream>>>(attn, colsum, s_p, s_s);
    if (ind < NUM_ITER_) {
      updates_kernel<<<B_ * S_, 64, 0, stream>>>(attn, colsum, v, updates);
      gru_mlp_kernel<<<B_ * S_, 64, 0, stream>>>(
          updates, gru_k, gru_rk, gru_b, nm_g, nm_b, mlp_w1, mlp_b1, mlp_w2,
          mlp_b2, slots);
    }
  }
  const int total_out = B_ * S_ * D_ + 2 * (B_ * S_ * 2) + B_ * N_ * S_;
  finalize_kernel<<<(total_out + 255) / 256, 256, 0, stream>>>(
      slots, s_p, s_s, attn, colsum, (float*)d_out);
}